// Encoder_77799037599895
// MI455X (gfx1250) — compile-verified
//
#include <hip/hip_runtime.h>
#include <math.h>

typedef float v2f __attribute__((ext_vector_type(2)));
typedef float v8f __attribute__((ext_vector_type(8)));

// D = A(16x4 f32) * B(4x16 f32) + C(16x16 f32), full fp32 WMMA on gfx1250
#define WMMA4(A, B, C) \
  __builtin_amdgcn_wmma_f32_16x16x4_f32(false, (A), false, (B), (short)0, (C), false, false)

// B fragment (4x16): lanes 0-15 hold rows k,k+1 of column (col0+lane);
// lanes 16-31 hold rows k+2,k+3. Caller passes k already offset by 2*(lane>>4).
__device__ __forceinline__ v2f load_b(const float* __restrict__ W, int ld, int k, int col) {
  v2f b;
  b.x = W[k * ld + col];
  b.y = W[(k + 1) * ld + col];
  return b;
}

__device__ __forceinline__ float sigmoid_f(float v) { return 1.0f / (1.0f + __expf(-v)); }

// ---------------- sparse preparation ----------------

__global__ void deg_kernel(const int* __restrict__ src, const float* __restrict__ ew,
                           float* __restrict__ deg, int E) {
  int e = blockIdx.x * blockDim.x + threadIdx.x;
  if (e < E) unsafeAtomicAdd(&deg[src[e]], ew[e]);
}

__global__ void dis_kernel(float* __restrict__ deg, int N) {
  int i = blockIdx.x * blockDim.x + threadIdx.x;
  if (i < N) {
    float d = deg[i];
    deg[i] = (d > 0.0f) ? (1.0f / sqrtf(d)) : 0.0f;   // in-place deg -> D^-1/2
  }
}

__global__ void norm_kernel(const int* __restrict__ src, const int* __restrict__ dst,
                            const float* __restrict__ ew, const float* __restrict__ dis,
                            float* __restrict__ nrm, int E) {
  int e = blockIdx.x * blockDim.x + threadIdx.x;
  if (e < E) nrm[e] = -dis[src[e]] * ew[e] * dis[dst[e]];
}

// ---------------- sparse L_hat @ V (scatter-add) ----------------
// One work-item per (edge, 4-float chunk). Two feature matrices per pass.

__global__ void lap2_kernel(const int* __restrict__ src, const int* __restrict__ dst,
                            const float* __restrict__ nrm,
                            const float* __restrict__ va, const float* __restrict__ vb,
                            float* __restrict__ oa, float* __restrict__ ob, int E) {
  int idx = blockIdx.x * blockDim.x + threadIdx.x;
  if (idx >= E * 64) return;
  int e = idx >> 6;
  int c = (idx & 63) << 2;
  int s = src[e], d = dst[e];
  float w = nrm[e];
  const float4 a4 = *(const float4*)(va + (long)s * 256 + c);
  const float4 b4 = *(const float4*)(vb + (long)s * 256 + c);
  float* pa = oa + (long)d * 256 + c;
  float* pb = ob + (long)d * 256 + c;
  unsafeAtomicAdd(pa + 0, w * a4.x);
  unsafeAtomicAdd(pa + 1, w * a4.y);
  unsafeAtomicAdd(pa + 2, w * a4.z);
  unsafeAtomicAdd(pa + 3, w * a4.w);
  unsafeAtomicAdd(pb + 0, w * b4.x);
  unsafeAtomicAdd(pb + 1, w * b4.y);
  unsafeAtomicAdd(pb + 2, w * b4.z);
  unsafeAtomicAdd(pb + 3, w * b4.w);
}

__global__ void lap1_kernel(const int* __restrict__ src, const int* __restrict__ dst,
                            const float* __restrict__ nrm,
                            const float* __restrict__ va, float* __restrict__ oa, int E) {
  int idx = blockIdx.x * blockDim.x + threadIdx.x;
  if (idx >= E * 64) return;
  int e = idx >> 6;
  int c = (idx & 63) << 2;
  int s = src[e], d = dst[e];
  float w = nrm[e];
  const float4 a4 = *(const float4*)(va + (long)s * 256 + c);
  float* pa = oa + (long)d * 256 + c;
  unsafeAtomicAdd(pa + 0, w * a4.x);
  unsafeAtomicAdd(pa + 1, w * a4.y);
  unsafeAtomicAdd(pa + 2, w * a4.z);
  unsafeAtomicAdd(pa + 3, w * a4.w);
}

// ---------------- stage 1: fused gate GEMMs ----------------
// Zpre = x@Wxz0 + Lx@Wxz1 + h@Whz0 + Lh@Whz1 (+bias) -> Zs = sigmoid
// Rpre likewise -> hr = h * sigmoid(Rpre)
// Hx   = x@Wxh0 + Lx@Wxh1 + bxh + bhh
// Block: 32 rows x 256 cols, 8 waves; wave covers 2 row-tiles x 2 col-tiles.

#define KC 64
#define LDA 68  // KC + 4 pad: bank-conflict-free for the 16x4 f32 A-fragment reads

__global__ void __launch_bounds__(256) gemm_gates(
    const float* __restrict__ x, const float* __restrict__ Lx,
    const float* __restrict__ h, const float* __restrict__ Lh,
    const float* __restrict__ Wxz, const float* __restrict__ Whz,
    const float* __restrict__ Wxr, const float* __restrict__ Whr,
    const float* __restrict__ Wxh,
    const float* __restrict__ bxz, const float* __restrict__ bhz,
    const float* __restrict__ bxr, const float* __restrict__ bhr,
    const float* __restrict__ bxh, const float* __restrict__ bhh,
    float* __restrict__ Zs, float* __restrict__ hr, float* __restrict__ Hx) {
  __shared__ __align__(16) float As[4][32][LDA];
  const int tid = threadIdx.x;
  const int lane = tid & 31;
  const int wave = tid >> 5;
  const int rowBase = blockIdx.x * 32;
  const int n = lane & 15;
  const int khalf = (lane >> 4) << 1;

  const float* Amats[4] = {x, Lx, h, Lh};

  v8f accZ[2][2] = {};
  v8f accR[2][2] = {};
  v8f accH[2][2] = {};

  for (int kc = 0; kc < 256; kc += KC) {
    __syncthreads();
#pragma unroll
    for (int m = 0; m < 4; ++m) {
      const float* srcp = Amats[m] + (long)rowBase * 256 + kc;
#pragma unroll
      for (int i = 0; i < 2; ++i) {
        int idx = tid + i * 256;          // 0..511 -> (row, float4-col)
        int r = idx >> 4;
        int c4 = (idx & 15) << 2;
        *(float4*)&As[m][r][c4] = *(const float4*)(srcp + (long)r * 256 + c4);
      }
    }
    __syncthreads();

    for (int kk = 0; kk < KC; kk += 4) {
      v2f a[4][2];
#pragma unroll
      for (int m = 0; m < 4; ++m)
#pragma unroll
        for (int rt = 0; rt < 2; ++rt)
          a[m][rt] = *(const v2f*)&As[m][rt * 16 + n][kk + khalf];

      const int kg = kc + kk + khalf;
#pragma unroll
      for (int ct = 0; ct < 2; ++ct) {
        const int col = wave * 32 + ct * 16 + n;
        v2f bZ0 = load_b(Wxz,         256, kg, col);
        v2f bZ1 = load_b(Wxz + 65536, 256, kg, col);
        v2f bZ2 = load_b(Whz,         256, kg, col);
        v2f bZ3 = load_b(Whz + 65536, 256, kg, col);
        v2f bR0 = load_b(Wxr,         256, kg, col);
        v2f bR1 = load_b(Wxr + 65536, 256, kg, col);
        v2f bR2 = load_b(Whr,         256, kg, col);
        v2f bR3 = load_b(Whr + 65536, 256, kg, col);
        v2f bH0 = load_b(Wxh,         256, kg, col);
        v2f bH1 = load_b(Wxh + 65536, 256, kg, col);
#pragma unroll
        for (int rt = 0; rt < 2; ++rt) {
          accZ[rt][ct] = WMMA4(a[0][rt], bZ0, accZ[rt][ct]);
          accZ[rt][ct] = WMMA4(a[1][rt], bZ1, accZ[rt][ct]);
          accZ[rt][ct] = WMMA4(a[2][rt], bZ2, accZ[rt][ct]);
          accZ[rt][ct] = WMMA4(a[3][rt], bZ3, accZ[rt][ct]);
          accR[rt][ct] = WMMA4(a[0][rt], bR0, accR[rt][ct]);
          accR[rt][ct] = WMMA4(a[1][rt], bR1, accR[rt][ct]);
          accR[rt][ct] = WMMA4(a[2][rt], bR2, accR[rt][ct]);
          accR[rt][ct] = WMMA4(a[3][rt], bR3, accR[rt][ct]);
          accH[rt][ct] = WMMA4(a[0][rt], bH0, accH[rt][ct]);
          accH[rt][ct] = WMMA4(a[1][rt], bH1, accH[rt][ct]);
        }
      }
    }
  }

  // Epilogue: C/D layout -> element (M = i + 8*(lane>>4), N = lane&15)
  const int mh = (lane >> 4) << 3;
#pragma unroll
  for (int rt = 0; rt < 2; ++rt)
#pragma unroll
    for (int ct = 0; ct < 2; ++ct) {
      const int col = wave * 32 + ct * 16 + n;
      const float bz = bxz[col] + bhz[col];
      const float br = bxr[col] + bhr[col];
      const float bh = bxh[col] + bhh[col];
#pragma unroll
      for (int i = 0; i < 8; ++i) {
        const long o = (long)(rowBase + rt * 16 + mh + i) * 256 + col;
        const float z = sigmoid_f(accZ[rt][ct][i] + bz);
        Zs[o] = z;
        const float r = sigmoid_f(accR[rt][ct][i] + br);
        hr[o] = h[o] * r;
        Hx[o] = accH[rt][ct][i] + bh;
      }
    }
}

// ---------------- stage 2: Ht GEMM + GRU blend ----------------
// Htpre = hr@Whh0 + Lhr@Whh1 + Hx ; h0 = Z*h + (1-Z)*tanh(Htpre)

__global__ void __launch_bounds__(256) gemm_ht(
    const float* __restrict__ hr, const float* __restrict__ Lhr,
    const float* __restrict__ Whh,
    const float* __restrict__ Hx, const float* __restrict__ Zs,
    const float* __restrict__ h,
    float* __restrict__ h0out, float* __restrict__ rh0) {
  __shared__ __align__(16) float As[2][32][LDA];
  const int tid = threadIdx.x;
  const int lane = tid & 31;
  const int wave = tid >> 5;
  const int rowBase = blockIdx.x * 32;
  const int n = lane & 15;
  const int khalf = (lane >> 4) << 1;

  const float* Amats[2] = {hr, Lhr};
  v8f acc[2][2] = {};

  for (int kc = 0; kc < 256; kc += KC) {
    __syncthreads();
#pragma unroll
    for (int m = 0; m < 2; ++m) {
      const float* srcp = Amats[m] + (long)rowBase * 256 + kc;
#pragma unroll
      for (int i = 0; i < 2; ++i) {
        int idx = tid + i * 256;
        int r = idx >> 4;
        int c4 = (idx & 15) << 2;
        *(float4*)&As[m][r][c4] = *(const float4*)(srcp + (long)r * 256 + c4);
      }
    }
    __syncthreads();

    for (int kk = 0; kk < KC; kk += 4) {
      v2f a[2][2];
#pragma unroll
      for (int m = 0; m < 2; ++m)
#pragma unroll
        for (int rt = 0; rt < 2; ++rt)
          a[m][rt] = *(const v2f*)&As[m][rt * 16 + n][kk + khalf];

      const int kg = kc + kk + khalf;
#pragma unroll
      for (int ct = 0; ct < 2; ++ct) {
        const int col = wave * 32 + ct * 16 + n;
        v2f b0 = load_b(Whh,         256, kg, col);
        v2f b1 = load_b(Whh + 65536, 256, kg, col);
#pragma unroll
        for (int rt = 0; rt < 2; ++rt) {
          acc[rt][ct] = WMMA4(a[0][rt], b0, acc[rt][ct]);
          acc[rt][ct] = WMMA4(a[1][rt], b1, acc[rt][ct]);
        }
      }
    }
  }

  const int mh = (lane >> 4) << 3;
#pragma unroll
  for (int rt = 0; rt < 2; ++rt)
#pragma unroll
    for (int ct = 0; ct < 2; ++ct) {
      const int col = wave * 32 + ct * 16 + n;
#pragma unroll
      for (int i = 0; i < 8; ++i) {
        const long o = (long)(rowBase + rt * 16 + mh + i) * 256 + col;
        const float ht = tanhf(acc[rt][ct][i] + Hx[o]);
        const float z = Zs[o];
        const float h0 = z * h[o] + (1.0f - z) * ht;
        h0out[o] = h0;
        rh0[o] = fmaxf(h0, 0.0f);
      }
    }
}

// ---------------- stage 3: out = relu(h0) @ Wl + bl ----------------

__global__ void __launch_bounds__(256) gemm_out(
    const float* __restrict__ rh0, const float* __restrict__ Wl,
    const float* __restrict__ bl, float* __restrict__ outp) {
  __shared__ __align__(16) float As[32][LDA];
  const int tid = threadIdx.x;
  const int lane = tid & 31;
  const int wave = tid >> 5;           // 8 waves x 16 cols = 128 cols
  const int rowBase = blockIdx.x * 32;
  const int n = lane & 15;
  const int khalf = (lane >> 4) << 1;

  v8f acc[2] = {};

  for (int kc = 0; kc < 256; kc += KC) {
    __syncthreads();
    const float* srcp = rh0 + (long)rowBase * 256 + kc;
#pragma unroll
    for (int i = 0; i < 2; ++i) {
      int idx = tid + i * 256;
      int r = idx >> 4;
      int c4 = (idx & 15) << 2;
      *(float4*)&As[r][c4] = *(const float4*)(srcp + (long)r * 256 + c4);
    }
    __syncthreads();

    for (int kk = 0; kk < KC; kk += 4) {
      const int kg = kc + kk + khalf;
      const int col = wave * 16 + n;
      v2f b0 = load_b(Wl, 128, kg, col);
#pragma unroll
      for (int rt = 0; rt < 2; ++rt) {
        v2f a = *(const v2f*)&As[rt * 16 + n][kk + khalf];
        acc[rt] = WMMA4(a, b0, acc[rt]);
      }
    }
  }

  const int mh = (lane >> 4) << 3;
  const int col = wave * 16 + n;
  const float bb = bl[col];
#pragma unroll
  for (int rt = 0; rt < 2; ++rt)
#pragma unroll
    for (int i = 0; i < 8; ++i)
      outp[(long)(rowBase + rt * 16 + mh + i) * 128 + col] = acc[rt][i] + bb;
}

// ---------------- host orchestration ----------------

extern "C" void kernel_launch(void* const* d_in, const int* in_sizes, int n_in,
                              void* d_out, int out_size, void* d_ws, size_t ws_size,
                              hipStream_t stream) {
  (void)n_in; (void)out_size; (void)ws_size;
  const float* x   = (const float*)d_in[0];
  const float* h   = (const float*)d_in[1];
  const int*   ei  = (const int*)d_in[2];
  const float* ew  = (const float*)d_in[3];
  const float* Wxz = (const float*)d_in[4];  const float* bxz = (const float*)d_in[5];
  const float* Whz = (const float*)d_in[6];  const float* bhz = (const float*)d_in[7];
  const float* Wxr = (const float*)d_in[8];  const float* bxr = (const float*)d_in[9];
  const float* Whr = (const float*)d_in[10]; const float* bhr = (const float*)d_in[11];
  const float* Wxh = (const float*)d_in[12]; const float* bxh = (const float*)d_in[13];
  const float* Whh = (const float*)d_in[14]; const float* bhh = (const float*)d_in[15];
  const float* Wl  = (const float*)d_in[16]; const float* bl  = (const float*)d_in[17];

  const int N = in_sizes[0] / 256;   // 20000 (divisible by 32)
  const int E = in_sizes[3];         // 320000
  const int* src = ei;
  const int* dst = ei + E;

  float* ws = (float*)d_ws;
  size_t o = 0;
  float* deg = ws + o; o += ((size_t)N + 511) / 512 * 512;
  float* nrm = ws + o; o += ((size_t)E + 511) / 512 * 512;
  float* Lx  = ws + o; o += (size_t)N * 256;
  float* Lh  = ws + o; o += (size_t)N * 256;
  float* Zs  = ws + o; o += (size_t)N * 256;
  float* hrb = ws + o; o += (size_t)N * 256;
  float* Hx  = ws + o; o += (size_t)N * 256;
  float* Lhr = Lx;   // Lx dead after gemm_gates
  float* rh0 = Lh;   // Lh dead after gemm_gates

  float* outp = (float*)d_out;
  float* h0p  = outp + (size_t)N * 128;

  const int lapWork = E * 64;

  hipMemsetAsync(deg, 0, (size_t)N * sizeof(float), stream);
  deg_kernel<<<(E + 255) / 256, 256, 0, stream>>>(src, ew, deg, E);
  dis_kernel<<<(N + 255) / 256, 256, 0, stream>>>(deg, N);
  norm_kernel<<<(E + 255) / 256, 256, 0, stream>>>(src, dst, ew, deg, nrm, E);

  hipMemsetAsync(Lx, 0, (size_t)N * 256 * sizeof(float), stream);
  hipMemsetAsync(Lh, 0, (size_t)N * 256 * sizeof(float), stream);
  lap2_kernel<<<(lapWork + 255) / 256, 256, 0, stream>>>(src, dst, nrm, x, h, Lx, Lh, E);

  gemm_gates<<<N / 32, 256, 0, stream>>>(x, Lx, h, Lh, Wxz, Whz, Wxr, Whr, Wxh,
                                         bxz, bhz, bxr, bhr, bxh, bhh, Zs, hrb, Hx);

  hipMemsetAsync(Lhr, 0, (size_t)N * 256 * sizeof(float), stream);
  lap1_kernel<<<(lapWork + 255) / 256, 256, 0, stream>>>(src, dst, nrm, hrb, Lhr, E);

  gemm_ht<<<N / 32, 256, 0, stream>>>(hrb, Lhr, Whh, Hx, Zs, h, h0p, rh0);

  gemm_out<<<N / 32, 256, 0, stream>>>(rh0, Wl, bl, outp);
}